// GATLayer_32710470927091
// MI455X (gfx1250) — compile-verified
//
#include <hip/hip_runtime.h>
#include <hip/hip_bf16.h>
#include <math.h>

// Problem constants (match reference)
constexpr int Dm  = 192;   // d_model
constexpr int DEe = 96;    // d_edge
constexpr int Hh  = 8;     // heads
constexpr int DHh = 24;    // d_head
constexpr int DFf = 384;   // ff hidden

typedef __attribute__((ext_vector_type(2))) float v2f;
typedef __attribute__((ext_vector_type(8))) float v8f;

// ---------------------------------------------------------------------------
// Wave-level fp32 WMMA GEMM: one wave computes a 16-row x (CT*16)-col tile,
// K in steps of 4 with V_WMMA_F32_16X16X4_F32. Fragment fetches return v2f
// (contiguous pairs -> global_load_b64). Out-of-range rows must be handled by
// clamping in afetch2 (unconditional loads keep EXEC all-ones for WMMA).
// ---------------------------------------------------------------------------
template <int CT, typename AF, typename BF, typename EP>
__device__ __forceinline__ void wave_gemm(int c0, int Kdim, AF afetch2, BF bfetch2, EP ep) {
  const int lane = threadIdx.x & 31;
  const int up   = lane >> 4;       // 0: K pair {0,1}, 1: K pair {2,3}
  const int l16  = lane & 15;       // M for A, N for B/C/D
  const int kp   = up * 2;
  v8f zero = {};
  v8f acc[CT];
#pragma unroll
  for (int t = 0; t < CT; ++t) acc[t] = zero;

  for (int k = 0; k < Kdim; k += 4) {
    v2f a = afetch2(l16, k + kp);
#pragma unroll
    for (int t = 0; t < CT; ++t) {
      v2f b = bfetch2(k + kp, c0 + t * 16 + l16);
      acc[t] = __builtin_amdgcn_wmma_f32_16x16x4_f32(
          false, a, false, b, (short)0, acc[t], false, false);
    }
  }
  const int rb = up * 8;            // C/D: VGPR i holds M = i (+8 for lanes>=16)
#pragma unroll
  for (int t = 0; t < CT; ++t)
#pragma unroll
    for (int i = 0; i < 8; ++i)
      ep(rb + i, c0 + t * 16 + l16, acc[t][i]);
}

__device__ __forceinline__ int clampRow(int r, int n) { return r < n ? r : n - 1; }

__device__ __forceinline__ void atomicMaxF(float* addr, float val) {
  unsigned int* ai = (unsigned int*)addr;
  unsigned int old = *ai;
  while (__uint_as_float(old) < val) {
    unsigned int assumed = old;
    old = atomicCAS(ai, assumed, __float_as_uint(val));
    if (old == assumed) break;
  }
}

// ---------------------------------------------------------------------------
// 0) weight transpose: out[c*R + r] = in[r*C + c]  (one-shot, ~1.5 MB total)
// ---------------------------------------------------------------------------
__global__ void k_transpose(const float* __restrict__ in, float* __restrict__ out,
                            int R, int C) {
  int i = blockIdx.x * blockDim.x + threadIdx.x;
  if (i >= R * C) return;
  int r = i / C, c = i % C;
  out[(size_t)c * R + r] = in[i];
}

// ---------------------------------------------------------------------------
// 1) init: mx = -inf, den = 0, num = 0
// ---------------------------------------------------------------------------
__global__ void k_init(float* __restrict__ mx, float* __restrict__ den,
                       float* __restrict__ num, long nH, long nNum) {
  long i = (long)blockIdx.x * blockDim.x + threadIdx.x;
  long stride = (long)gridDim.x * blockDim.x;
  for (long j = i; j < nNum; j += stride) num[j] = 0.f;
  for (long j = i; j < nH; j += stride) { mx[j] = -INFINITY; den[j] = 0.f; }
}

// ---------------------------------------------------------------------------
// 2) Q/K/V = x @ {Wq,Wk,Wv}   (WMMA, K=192; grid.y covers 576 cols / 64)
// ---------------------------------------------------------------------------
__global__ void k_qkv(const float* __restrict__ x,
                      const float* __restrict__ WqT, const float* __restrict__ WkT,
                      const float* __restrict__ WvT,
                      float* __restrict__ Q, float* __restrict__ K,
                      float* __restrict__ V, int n) {
  const int m0 = blockIdx.x * 16;
  const int c0 = blockIdx.y * 64;
  auto af = [&](int m, int k) -> v2f {
    int r = clampRow(m0 + m, n);
    return *(const v2f*)(x + (size_t)r * Dm + k);
  };
  auto bf = [&](int k, int c) -> v2f {
    const float* Wt = (c < Dm) ? WqT : (c < 2 * Dm) ? WkT : WvT;
    return *(const v2f*)(Wt + (size_t)(c % Dm) * Dm + k);
  };
  auto ep = [&](int mo, int c, float v) {
    int r = m0 + mo;
    if (r >= n) return;
    float* O = (c < Dm) ? Q : (c < 2 * Dm) ? K : V;
    O[(size_t)r * Dm + (c % Dm)] = v;
  };
  wave_gemm<4>(c0, Dm, af, bf, ep);
}

// ---------------------------------------------------------------------------
// 3) per-edge score + segment max (We in LDS, float4 row streams)
// ---------------------------------------------------------------------------
__global__ void k_score(const float* __restrict__ Q, const float* __restrict__ K,
                        const float* __restrict__ ef, const float* __restrict__ We,
                        const int* __restrict__ src, const int* __restrict__ dst,
                        float* __restrict__ score, float* __restrict__ mx, int e) {
  __shared__ float sWe[DEe * Hh];
  for (int i = threadIdx.x; i < DEe * Hh; i += blockDim.x) sWe[i] = We[i];
  __syncthreads();
  int ei = blockIdx.x * blockDim.x + threadIdx.x;
  if (ei >= e) return;
  const int s = src[ei], d = dst[ei];
  const float4* q4 = (const float4*)(Q + (size_t)d * Dm);
  const float4* k4 = (const float4*)(K + (size_t)s * Dm);
  float sc[Hh];
#pragma unroll
  for (int h = 0; h < Hh; ++h) {
    float a = 0.f;
#pragma unroll
    for (int j = 0; j < DHh / 4; ++j) {       // 6 float4 per head
      float4 qa = q4[h * (DHh / 4) + j];
      float4 kb = k4[h * (DHh / 4) + j];
      a += qa.x * kb.x + qa.y * kb.y + qa.z * kb.z + qa.w * kb.w;
    }
    sc[h] = a * 0.20412414523193154f;          // 1/sqrt(24)
  }
  const float4* er4 = (const float4*)(ef + (size_t)ei * DEe);
  for (int j4 = 0; j4 < DEe / 4; ++j4) {
    float4 f = er4[j4];
    const float* w0 = sWe + (j4 * 4) * Hh;
#pragma unroll
    for (int h = 0; h < Hh; ++h)
      sc[h] += f.x * w0[h] + f.y * w0[Hh + h] + f.z * w0[2 * Hh + h] + f.w * w0[3 * Hh + h];
  }
#pragma unroll
  for (int h = 0; h < Hh; ++h) {
    score[(size_t)ei * Hh + h] = sc[h];
    atomicMaxF(mx + (size_t)d * Hh + h, sc[h]);
  }
}

// ---------------------------------------------------------------------------
// 4) alpha = exp(score - mx[dst]); den += alpha; num += alpha * V[src]
// ---------------------------------------------------------------------------
__global__ void k_alpha(const float* __restrict__ score, const float* __restrict__ mx,
                        const float* __restrict__ V, const int* __restrict__ src,
                        const int* __restrict__ dst, float* __restrict__ den,
                        float* __restrict__ num, int e) {
  int ei = blockIdx.x * blockDim.x + threadIdx.x;
  if (ei >= e) return;
  const int s = src[ei], d = dst[ei];
  float a[Hh];
#pragma unroll
  for (int h = 0; h < Hh; ++h) {
    a[h] = expf(score[(size_t)ei * Hh + h] - mx[(size_t)d * Hh + h]);
    atomicAdd(den + (size_t)d * Hh + h, a[h]);
  }
  const float4* v4 = (const float4*)(V + (size_t)s * Dm);
  float* nr = num + (size_t)d * Dm;
#pragma unroll
  for (int h = 0; h < Hh; ++h)
#pragma unroll
    for (int j = 0; j < DHh / 4; ++j) {
      float4 vv = v4[h * (DHh / 4) + j];
      float* np = nr + h * DHh + j * 4;
      atomicAdd(np + 0, a[h] * vv.x);
      atomicAdd(np + 1, a[h] * vv.y);
      atomicAdd(np + 2, a[h] * vv.z);
      atomicAdd(np + 3, a[h] * vv.w);
    }
}

// ---------------------------------------------------------------------------
// 5) o = (num/(den+1e-8)) @ Wo + bo   (WMMA, agg formed in the A-fetch)
// ---------------------------------------------------------------------------
__global__ void k_oproj(const float* __restrict__ num, const float* __restrict__ den,
                        const float* __restrict__ WoT, const float* __restrict__ bo,
                        float* __restrict__ o, int n) {
  const int m0 = blockIdx.x * 16;
  const int c0 = blockIdx.y * 64;
  auto af = [&](int m, int k) -> v2f {
    int r = clampRow(m0 + m, n);
    v2f nu = *(const v2f*)(num + (size_t)r * Dm + k);
    float dd = den[(size_t)r * Hh + (k / DHh)] + 1e-8f;   // pair stays in one head (k even)
    nu.x /= dd;
    nu.y /= dd;
    return nu;
  };
  auto bf = [&](int k, int c) -> v2f {
    return *(const v2f*)(WoT + (size_t)c * Dm + k);
  };
  auto ep = [&](int mo, int c, float v) {
    int r = m0 + mo;
    if (r < n) o[(size_t)r * Dm + c] = v + bo[c];
  };
  wave_gemm<4>(c0, Dm, af, bf, ep);
}

// ---------------------------------------------------------------------------
// 6) g = sigmoid([o|x] @ Wg + bg); z1 = g*o + (1-g)*x   (WMMA, K=384)
// ---------------------------------------------------------------------------
__global__ void k_gate(const float* __restrict__ o, const float* __restrict__ x,
                       const float* __restrict__ WgT, const float* __restrict__ bg,
                       float* __restrict__ z1, int n) {
  const int m0 = blockIdx.x * 16;
  const int c0 = blockIdx.y * 64;
  auto af = [&](int m, int k) -> v2f {
    int r = clampRow(m0 + m, n);
    const float* base = (k < Dm) ? (o + (size_t)r * Dm + k)
                                 : (x + (size_t)r * Dm + (k - Dm));
    return *(const v2f*)base;    // pair never straddles the 192 boundary (k even)
  };
  auto bf = [&](int k, int c) -> v2f {
    return *(const v2f*)(WgT + (size_t)c * (2 * Dm) + k);
  };
  auto ep = [&](int mo, int c, float v) {
    int r = m0 + mo;
    if (r >= n) return;
    float g = 1.f / (1.f + expf(-(v + bg[c])));
    size_t idx = (size_t)r * Dm + c;
    z1[idx] = g * o[idx] + (1.f - g) * x[idx];
  };
  wave_gemm<4>(c0, 2 * Dm, af, bf, ep);
}

// ---------------------------------------------------------------------------
// 7/10) LayerNorm: one wave per row (192 = 32 lanes x 6), shfl reduction
// ---------------------------------------------------------------------------
__global__ void k_ln(const float* __restrict__ in, const float* __restrict__ sc,
                     const float* __restrict__ bi, float* __restrict__ out, int n) {
  int row = blockIdx.x * (blockDim.x >> 5) + (threadIdx.x >> 5);
  if (row >= n) return;
  int lane = threadIdx.x & 31;
  const float* r = in + (size_t)row * Dm;
  float v[6], s = 0.f, sq = 0.f;
#pragma unroll
  for (int t = 0; t < 6; ++t) {
    v[t] = r[t * 32 + lane];
    s += v[t];
    sq += v[t] * v[t];
  }
#pragma unroll
  for (int off = 16; off > 0; off >>= 1) {
    s += __shfl_xor(s, off);
    sq += __shfl_xor(sq, off);
  }
  float mean = s * (1.f / Dm);
  float var = sq * (1.f / Dm) - mean * mean;
  float rstd = rsqrtf(var + 1e-5f);
  float* orow = out + (size_t)row * Dm;
#pragma unroll
  for (int t = 0; t < 6; ++t) {
    int c = t * 32 + lane;
    orow[c] = (v[t] - mean) * rstd * sc[c] + bi[c];
  }
}

// ---------------------------------------------------------------------------
// 8) h1 = silu(y @ Wf1 + bf1)   (WMMA, 384 output cols)
// ---------------------------------------------------------------------------
__global__ void k_ff1(const float* __restrict__ y, const float* __restrict__ Wf1T,
                      const float* __restrict__ bf1, float* __restrict__ h1, int n) {
  const int m0 = blockIdx.x * 16;
  const int c0 = blockIdx.y * 64;
  auto af = [&](int m, int k) -> v2f {
    int r = clampRow(m0 + m, n);
    return *(const v2f*)(y + (size_t)r * Dm + k);
  };
  auto bf = [&](int k, int c) -> v2f {
    return *(const v2f*)(Wf1T + (size_t)c * Dm + k);
  };
  auto ep = [&](int mo, int c, float v) {
    int r = m0 + mo;
    if (r >= n) return;
    float u = v + bf1[c];
    h1[(size_t)r * DFf + c] = u / (1.f + expf(-u));  // silu
  };
  wave_gemm<4>(c0, Dm, af, bf, ep);
}

// ---------------------------------------------------------------------------
// 9) z2 = y + h1 @ Wf2 + bf2   (WMMA, K=384)
// ---------------------------------------------------------------------------
__global__ void k_ff2(const float* __restrict__ h1, const float* __restrict__ y,
                      const float* __restrict__ Wf2T, const float* __restrict__ bf2,
                      float* __restrict__ z2, int n) {
  const int m0 = blockIdx.x * 16;
  const int c0 = blockIdx.y * 64;
  auto af = [&](int m, int k) -> v2f {
    int r = clampRow(m0 + m, n);
    return *(const v2f*)(h1 + (size_t)r * DFf + k);
  };
  auto bf = [&](int k, int c) -> v2f {
    return *(const v2f*)(Wf2T + (size_t)c * DFf + k);
  };
  auto ep = [&](int mo, int c, float v) {
    int r = m0 + mo;
    if (r >= n) return;
    size_t idx = (size_t)r * Dm + c;
    z2[idx] = y[idx] + v + bf2[c];
  };
  wave_gemm<4>(c0, DFf, af, bf, ep);
}

// ---------------------------------------------------------------------------
extern "C" void kernel_launch(void* const* d_in, const int* in_sizes, int n_in,
                              void* d_out, int out_size, void* d_ws, size_t ws_size,
                              hipStream_t stream) {
  const float* x    = (const float*)d_in[0];
  const float* ef   = (const float*)d_in[1];
  const int*   eidx = (const int*)d_in[2];   // int32 (JAX x64 disabled)
  const float* Wq   = (const float*)d_in[3];
  const float* Wk   = (const float*)d_in[4];
  const float* Wv   = (const float*)d_in[5];
  const float* We   = (const float*)d_in[6];
  const float* Wo   = (const float*)d_in[7];
  const float* bo   = (const float*)d_in[8];
  const float* Wg   = (const float*)d_in[9];
  const float* bg   = (const float*)d_in[10];
  const float* l1s  = (const float*)d_in[11];
  const float* l1b  = (const float*)d_in[12];
  const float* Wf1  = (const float*)d_in[13];
  const float* bf1  = (const float*)d_in[14];
  const float* Wf2  = (const float*)d_in[15];
  const float* bf2  = (const float*)d_in[16];
  const float* l2s  = (const float*)d_in[17];
  const float* l2b  = (const float*)d_in[18];

  const int n = in_sizes[0] / Dm;   // 25000
  const int e = in_sizes[1] / DEe;  // 400000
  const int* src = eidx;
  const int* dst = eidx + e;

  // Workspace layout (floats); later stages reuse dead regions.
  float* ws = (float*)d_ws;
  const size_t NF = (size_t)n * Dm;
  float* Q     = ws;                 // -> reused as o
  float* K     = ws + NF;            // -> reused as z1
  float* V     = ws + 2 * NF;        // -> reused as y
  float* num   = ws + 3 * NF;        // -> reused as z2
  float* h1    = ws + 4 * NF;        // N x 384 (2*NF)
  float* score = ws + 6 * NF;        // E x 8
  float* mx    = score + (size_t)e * Hh;
  float* den   = mx + (size_t)n * Hh;
  // transposed weights (tail of workspace, ~1.5 MB)
  float* WqT  = den + (size_t)n * Hh;            // 192*192
  float* WkT  = WqT + Dm * Dm;
  float* WvT  = WkT + Dm * Dm;
  float* WoT  = WvT + Dm * Dm;
  float* WgT  = WoT + Dm * Dm;                   // [192][384]
  float* Wf1T = WgT + Dm * 2 * Dm;               // [384][192]
  float* Wf2T = Wf1T + DFf * Dm;                 // [192][384]
  float* o  = Q;
  float* z1 = K;
  float* y  = V;
  float* z2 = num;

  const int mt = (n + 15) / 16;      // 16-row tiles
  const dim3 wave(32);

  // 0) one-shot weight transposes (independent of graph inputs' values)
  const int sq = Dm * Dm, rc = Dm * DFf;
  k_transpose<<<(sq + 255) / 256, 256, 0, stream>>>(Wq, WqT, Dm, Dm);
  k_transpose<<<(sq + 255) / 256, 256, 0, stream>>>(Wk, WkT, Dm, Dm);
  k_transpose<<<(sq + 255) / 256, 256, 0, stream>>>(Wv, WvT, Dm, Dm);
  k_transpose<<<(sq + 255) / 256, 256, 0, stream>>>(Wo, WoT, Dm, Dm);
  k_transpose<<<(rc + 255) / 256, 256, 0, stream>>>(Wg, WgT, 2 * Dm, Dm);
  k_transpose<<<(rc + 255) / 256, 256, 0, stream>>>(Wf1, Wf1T, Dm, DFf);
  k_transpose<<<(rc + 255) / 256, 256, 0, stream>>>(Wf2, Wf2T, DFf, Dm);

  // 1) init accumulators
  k_init<<<2048, 256, 0, stream>>>(mx, den, num, (long)n * Hh, (long)n * Dm);

  // 2) Q/K/V projection (576 cols -> 9 groups of 64)
  k_qkv<<<dim3(mt, 9), wave, 0, stream>>>(x, WqT, WkT, WvT, Q, K, V, n);

  // 3) scores + segment max
  k_score<<<(e + 255) / 256, 256, 0, stream>>>(Q, K, ef, We, src, dst, score, mx, e);

  // 4) softmax numerator/denominator accumulation
  k_alpha<<<(e + 255) / 256, 256, 0, stream>>>(score, mx, V, src, dst, den, num, e);

  // 5) output projection
  k_oproj<<<dim3(mt, 3), wave, 0, stream>>>(num, den, WoT, bo, o, n);

  // 6) gated residual
  k_gate<<<dim3(mt, 3), wave, 0, stream>>>(o, x, WgT, bg, z1, n);

  // 7) layernorm 1
  k_ln<<<(n + 7) / 8, 256, 0, stream>>>(z1, l1s, l1b, y, n);

  // 8) FF up + silu (384 cols -> 6 groups)
  k_ff1<<<dim3(mt, 6), wave, 0, stream>>>(y, Wf1T, bf1, h1, n);

  // 9) FF down + residual
  k_ff2<<<dim3(mt, 3), wave, 0, stream>>>(h1, y, Wf2T, bf2, z2, n);

  // 10) layernorm 2 -> output
  k_ln<<<(n + 7) / 8, 256, 0, stream>>>(z2, l2s, l2b, (float*)d_out, n);
}